// NEWMYIN_57543971832337
// MI455X (gfx1250) — compile-verified
//
#include <hip/hip_runtime.h>
#include <hip/hip_bf16.h>

#define BB   2
#define CC   64
#define HH   256
#define WW   256
#define SZ   8
#define STR  4
#define NUMS 63
#define LL   3969          /* NUMS*NUMS */
#define LP   3984          /* LL padded to 16*249 (= 48*83) */
#define KK2  64
#define CK   4096          /* CC*KK2 */
#define NTB  125           /* 2*NUMS-1 */
#define EPSF 1e-5f
#define NEGINF (-1.0e30f)

typedef float v2f __attribute__((ext_vector_type(2)));
typedef float v8f __attribute__((ext_vector_type(8)));

__device__ __forceinline__ v8f wmma_f32(v2f a, v2f b, v8f c) {
  // V_WMMA_F32_16X16X4_F32 : D(16x16 f32) = A(16x4 f32) * B(4x16 f32) + C
  return __builtin_amdgcn_wmma_f32_16x16x4_f32(false, a, false, b, (short)0, c,
                                               false, false);
}

__device__ __forceinline__ float hmax16(float v) {
  v = fmaxf(v, __shfl_xor(v, 1, 32));
  v = fmaxf(v, __shfl_xor(v, 2, 32));
  v = fmaxf(v, __shfl_xor(v, 4, 32));
  v = fmaxf(v, __shfl_xor(v, 8, 32));
  return v;
}
__device__ __forceinline__ float hsum16(float v) {
  v += __shfl_xor(v, 1, 32);
  v += __shfl_xor(v, 2, 32);
  v += __shfl_xor(v, 4, 32);
  v += __shfl_xor(v, 8, 32);
  return v;
}

// ---------------------------------------------------------------------------
// K1: per-patch masked stats + instance norm.
// One thread per (b,l). Writes in_f/in_b as [B][CK][LP] (row = c*64+k, col = l)
// so consecutive threads (consecutive l) store coalesced. Also writes bg
// variance sb [B][CC][LP]. Padded columns l>=LL are zero-filled.
// ---------------------------------------------------------------------------
__global__ void k_stats(const float* __restrict__ x, const float* __restrict__ mask,
                        float* __restrict__ in_f, float* __restrict__ in_b,
                        float* __restrict__ sb_out) {
  int t = blockIdx.x * blockDim.x + threadIdx.x;
  int b = t / LP;
  int l = t - b * LP;
  if (b >= BB) return;
  float* inf_b = in_f + (size_t)b * CK * LP;
  float* inb_b = in_b + (size_t)b * CK * LP;
  float* sb_b  = sb_out + (size_t)b * CC * LP;
  if (l >= LL) {  // zero padded columns
    for (int ck = 0; ck < CK; ++ck) {
      size_t idx = (size_t)ck * LP + l;
      inf_b[idx] = 0.0f; inb_b[idx] = 0.0f;
    }
    for (int c = 0; c < CC; ++c) sb_b[(size_t)c * LP + l] = 0.0f;
    return;
  }
  int lh = l / NUMS, lw = l - lh * NUMS;
  int i0 = lh * STR, j0 = lw * STR;
  const float* mrow = mask + (size_t)b * HH * WW;
  unsigned long long bits = 0ull;
  #pragma unroll
  for (int kh = 0; kh < SZ; ++kh)
    #pragma unroll
    for (int kw = 0; kw < SZ; ++kw)
      if (mrow[(i0 + kh) * WW + j0 + kw] > 0.5f)
        bits |= 1ull << (kh * SZ + kw);
  float numf = (float)__popcll(bits);
  float rnf = 1.0f / (numf + EPSF);
  float rnb = 1.0f / ((64.0f - numf) + EPSF);

  for (int c = 0; c < CC; ++c) {
    const float* xrow = x + ((size_t)b * CC + c) * HH * WW;
    float v[KK2];
    float sf = 0.0f, sg = 0.0f;
    #pragma unroll
    for (int k = 0; k < KK2; ++k) {
      int kh = k >> 3, kw = k & 7;
      float xv = xrow[(i0 + kh) * WW + j0 + kw];
      v[k] = xv;
      bool m1 = (bits >> k) & 1ull;
      sf += m1 ? xv : 0.0f;
      sg += m1 ? 0.0f : xv;
    }
    float muf = sf * rnf, mub = sg * rnb;
    float varf = 0.0f, varb = 0.0f;
    #pragma unroll
    for (int k = 0; k < KK2; ++k) {
      bool m1 = (bits >> k) & 1ull;
      float df = m1 ? (v[k] - muf) : 0.0f;
      float db = m1 ? 0.0f : (v[k] - mub);
      varf += df * df; varb += db * db;
    }
    varf *= rnf; varb *= rnb;
    float rvf = 1.0f / varf, rvb = 1.0f / varb;
    #pragma unroll
    for (int k = 0; k < KK2; ++k) {
      bool m1 = (bits >> k) & 1ull;
      size_t idx = (size_t)(c * KK2 + k) * LP + l;
      inf_b[idx] = m1 ? (v[k] - muf) * rvf : muf;        // in_f
      inb_b[idx] = m1 ? mub : (v[k] - mub) * rvb;        // in_b
    }
    sb_b[(size_t)c * LP + l] = varb;
  }
}

// ---------------------------------------------------------------------------
// K2: token GEMM  OUT[b][64][LP] = Wm[64 x 4096] @ IN[b][4096 x LP]
// One wave per 16x48 output tile: 3 independent WMMA accumulator chains
// (pipelines the f32 WMMA RAW hazard) and 3x reuse of each A fragment.
// B operand is streamed with global_prefetch 32 K-rows ahead.
// ---------------------------------------------------------------------------
__global__ void k_tok(const float* __restrict__ Wm, const float* __restrict__ IN,
                      float* __restrict__ OUT) {
  int tile = blockIdx.x;
  int nt = tile % (LP / 48); tile /= (LP / 48);
  int mt = tile % (CC / 16);
  int b  = tile / (CC / 16);
  int lane = threadIdx.x;
  int half = lane >> 4;
  int lid  = lane & 15;
  int m0 = mt * 16, n0 = nt * 48;
  const float* inb = IN + (size_t)b * CK * LP;
  v8f acc0 = {}, acc1 = {}, acc2 = {};
  #pragma unroll 2
  for (int k0 = 0; k0 < CK; k0 += 4) {
    int ka = k0 + half * 2;
    v2f a, b0, b1, b2;
    a.x  = Wm[(size_t)(m0 + lid) * CK + ka];
    a.y  = Wm[(size_t)(m0 + lid) * CK + ka + 1];
    const float* r0 = inb + (size_t)ka * LP + n0 + lid;
    const float* r1 = inb + (size_t)(ka + 1) * LP + n0 + lid;
    b0.x = r0[0];  b0.y = r1[0];
    b1.x = r0[16]; b1.y = r1[16];
    b2.x = r0[32]; b2.y = r1[32];
    if (k0 + 32 < CK)
      __builtin_prefetch(r0 + (size_t)32 * LP, 0, 1);   // global_prefetch_b8
    acc0 = wmma_f32(a, b0, acc0);
    acc1 = wmma_f32(a, b1, acc1);
    acc2 = wmma_f32(a, b2, acc2);
  }
  float* outb = OUT + (size_t)b * CC * LP;
  #pragma unroll
  for (int i = 0; i < 8; ++i) {
    int m = m0 + i + half * 8;                 // C/D layout: row = vgpr + 8*half
    float* orow = outb + (size_t)m * LP + n0 + lid;
    orow[0]  = acc0[i];                        // col = lane%16 (+0/16/32)
    orow[16] = acc1[i];
    orow[32] = acc2[i];
  }
}

// ---------------------------------------------------------------------------
// K3: attention + new_std, flash-style two passes, one wave per 16-row block.
//   logits[l,m] = sum_c tokf[c,l]*tokb[c,m] + rpb[rpi(l,m)]
//   new_std[c,l] = sum_m softmax_m(logits[l,:]) * sb[c,m]
// QK^T uses two interleaved accumulator chains; P@V has 4 independent chains.
// ---------------------------------------------------------------------------
__global__ void k_attn(const float* __restrict__ tokf, const float* __restrict__ tokb,
                       const float* __restrict__ sbv, const float* __restrict__ rpb,
                       float* __restrict__ newstd) {
  __shared__ float Plds[16][17];
  int blk = blockIdx.x;
  int lt = blk % (LP / 16);
  int b  = blk / (LP / 16);
  int lane = threadIdx.x, half = lane >> 4, lid = lane & 15;
  int l0 = lt * 16;
  const float* qf = tokf + (size_t)b * CC * LP;
  const float* kb = tokb + (size_t)b * CC * LP;
  const float* vb = sbv  + (size_t)b * CC * LP;

  // cache Q fragments (A-layout): element (M=lid, K=ks*4 + 2*half (+1))
  v2f aq[16];
  #pragma unroll
  for (int ks = 0; ks < 16; ++ks) {
    int ka = ks * 4 + half * 2;
    aq[ks].x = qf[(size_t)ka * LP + l0 + lid];
    aq[ks].y = qf[(size_t)(ka + 1) * LP + l0 + lid];
  }

  float rmax[8], rsum[8];
  #pragma unroll
  for (int i = 0; i < 8; ++i) { rmax[i] = NEGINF; rsum[i] = 0.0f; }

  // ---- pass 1: online row max / sum-exp ----
  for (int mt = 0; mt < LP / 16; ++mt) {
    int m0 = mt * 16;
    v8f lg0 = {}, lg1 = {};
    #pragma unroll
    for (int ks = 0; ks < 8; ++ks) {
      int ka0 = (2 * ks) * 4 + half * 2;
      int ka1 = (2 * ks + 1) * 4 + half * 2;
      v2f b0, b1;
      b0.x = kb[(size_t)ka0 * LP + m0 + lid];
      b0.y = kb[(size_t)(ka0 + 1) * LP + m0 + lid];
      b1.x = kb[(size_t)ka1 * LP + m0 + lid];
      b1.y = kb[(size_t)(ka1 + 1) * LP + m0 + lid];
      lg0 = wmma_f32(aq[2 * ks], b0, lg0);
      lg1 = wmma_f32(aq[2 * ks + 1], b1, lg1);
    }
    v8f lg = lg0 + lg1;
    int m = m0 + lid;
    bool mvalid = (m < LL);
    int mh = m / NUMS, mw = m - mh * NUMS;
    #pragma unroll
    for (int i = 0; i < 8; ++i) {
      int l = l0 + i + half * 8;
      float val = NEGINF;
      if (mvalid && l < LL) {
        int lh2 = l / NUMS, lw2 = l - lh2 * NUMS;
        int rpi = (lh2 - mh + (NUMS - 1)) * NTB + (lw2 - mw + (NUMS - 1));
        val = lg[i] + rpb[rpi];
      }
      float tmax = hmax16(val);
      float tsum = hsum16(__expf(val - tmax));
      float nm = fmaxf(rmax[i], tmax);
      rsum[i] = rsum[i] * __expf(rmax[i] - nm) + tsum * __expf(tmax - nm);
      rmax[i] = nm;
    }
  }
  float rinv[8];
  #pragma unroll
  for (int i = 0; i < 8; ++i) rinv[i] = 1.0f / rsum[i];

  // ---- pass 2: P = exp(logit - rowmax); acc += P @ V  (V = sb^T) ----
  v8f oc[4];
  #pragma unroll
  for (int t = 0; t < 4; ++t) oc[t] = (v8f){};
  for (int mt = 0; mt < LP / 16; ++mt) {
    int m0 = mt * 16;
    v8f lg0 = {}, lg1 = {};
    #pragma unroll
    for (int ks = 0; ks < 8; ++ks) {
      int ka0 = (2 * ks) * 4 + half * 2;
      int ka1 = (2 * ks + 1) * 4 + half * 2;
      v2f b0, b1;
      b0.x = kb[(size_t)ka0 * LP + m0 + lid];
      b0.y = kb[(size_t)(ka0 + 1) * LP + m0 + lid];
      b1.x = kb[(size_t)ka1 * LP + m0 + lid];
      b1.y = kb[(size_t)(ka1 + 1) * LP + m0 + lid];
      lg0 = wmma_f32(aq[2 * ks], b0, lg0);
      lg1 = wmma_f32(aq[2 * ks + 1], b1, lg1);
    }
    v8f lg = lg0 + lg1;
    int m = m0 + lid;
    bool mvalid = (m < LL);
    int mh = m / NUMS, mw = m - mh * NUMS;
    #pragma unroll
    for (int i = 0; i < 8; ++i) {
      int l = l0 + i + half * 8;
      float p = 0.0f;
      if (mvalid && l < LL) {
        int lh2 = l / NUMS, lw2 = l - lh2 * NUMS;
        int rpi = (lh2 - mh + (NUMS - 1)) * NTB + (lw2 - mw + (NUMS - 1));
        p = __expf(lg[i] + rpb[rpi] - rmax[i]);
      }
      Plds[i + half * 8][lid] = p;   // D-layout -> LDS (row l-in-tile, col m-in-tile)
    }
    __syncthreads();                 // single wave per block: wave-local sync
    #pragma unroll
    for (int ks = 0; ks < 4; ++ks) {
      int ka = ks * 4 + half * 2;
      v2f ap;                        // P as A: element (M=lid, K=ka)
      ap.x = Plds[lid][ka];
      ap.y = Plds[lid][ka + 1];
      #pragma unroll
      for (int ct = 0; ct < 4; ++ct) {
        int c0 = ct * 16;
        v2f bv;                      // V as B: element (K=ka, N=lid) = sb[c0+lid][m0+ka]
        bv.x = vb[(size_t)(c0 + lid) * LP + m0 + ka];
        bv.y = vb[(size_t)(c0 + lid) * LP + m0 + ka + 1];
        oc[ct] = wmma_f32(ap, bv, oc[ct]);
      }
    }
    __syncthreads();
  }
  // store new_std [b][c][LP]; D-layout: row l = l0 + i + 8*half, col c = c0+lid
  float* nout = newstd + (size_t)b * CC * LP;
  #pragma unroll
  for (int ct = 0; ct < 4; ++ct)
    #pragma unroll
    for (int i = 0; i < 8; ++i) {
      int l = l0 + i + half * 8;
      if (l < LL)
        nout[(size_t)(ct * 16 + lid) * LP + l] = oc[ct][i] * rinv[i];
    }
}

// ---------------------------------------------------------------------------
// K4: gather-based fold + final blend.
// new_pf = (in_f + 1) * new_std ; fold /= coverage ; fg affine * mask + unorm_bg
// ---------------------------------------------------------------------------
__global__ void k_fold(const float* __restrict__ x, const float* __restrict__ mask,
                       const float* __restrict__ fg_g, const float* __restrict__ fg_b,
                       const float* __restrict__ bg_g, const float* __restrict__ bg_b,
                       const float* __restrict__ in_f, const float* __restrict__ newstd,
                       float* __restrict__ out) {
  size_t t = (size_t)blockIdx.x * blockDim.x + threadIdx.x;
  if (t >= (size_t)BB * CC * HH * WW) return;
  int j = (int)(t % WW); size_t r = t / WW;
  int i = (int)(r % HH); r /= HH;
  int c = (int)(r % CC);
  int b = (int)(r / CC);
  int lhmin = (i >= SZ - 1) ? ((i - (SZ - 1) + STR - 1) / STR) : 0;
  int lhmax = min(NUMS - 1, i / STR);
  int lwmin = (j >= SZ - 1) ? ((j - (SZ - 1) + STR - 1) / STR) : 0;
  int lwmax = min(NUMS - 1, j / STR);
  const float* infb = in_f + (size_t)b * CK * LP;
  const float* nsb  = newstd + (size_t)b * CC * LP;
  float sum = 0.0f; int cnt = 0;
  for (int lh = lhmin; lh <= lhmax; ++lh) {
    int kh = i - lh * STR;
    for (int lw = lwmin; lw <= lwmax; ++lw) {
      int kw = j - lw * STR;
      int l = lh * NUMS + lw;
      int k = kh * SZ + kw;
      float ns = nsb[(size_t)c * LP + l];
      sum += (infb[(size_t)(c * KK2 + k) * LP + l] + 1.0f) * ns;
      ++cnt;
    }
  }
  float val = sum / (float)cnt;
  float m = mask[(size_t)b * HH * WW + (size_t)i * WW + j];
  float xv = x[t];
  float invm = 1.0f - m;
  float fg  = (val * (1.0f + fg_g[c]) + fg_b[c]) * m;
  float ubg = ((xv * invm) * (1.0f + bg_g[c]) + bg_b[c]) * invm;
  out[t] = fg + ubg;
}

// ---------------------------------------------------------------------------
extern "C" void kernel_launch(void* const* d_in, const int* in_sizes, int n_in,
                              void* d_out, int out_size, void* d_ws, size_t ws_size,
                              hipStream_t stream) {
  (void)in_sizes; (void)n_in; (void)out_size; (void)ws_size;
  const float* x        = (const float*)d_in[0];
  const float* mask     = (const float*)d_in[1];
  const float* fg_gamma = (const float*)d_in[2];
  const float* fg_beta  = (const float*)d_in[3];
  const float* bg_gamma = (const float*)d_in[4];
  const float* bg_beta  = (const float*)d_in[5];
  const float* w_fore   = (const float*)d_in[6];
  const float* w_back   = (const float*)d_in[7];
  const float* rpb      = (const float*)d_in[8];
  float* out = (float*)d_out;

  float* ws   = (float*)d_ws;
  size_t nIN  = (size_t)BB * CK * LP;   // 32,636,928 floats each
  size_t nCL  = (size_t)BB * CC * LP;   //    509,952 floats each
  float* in_f = ws;
  float* in_b = in_f + nIN;
  float* sbv  = in_b + nIN;
  float* tokf = sbv + nCL;
  float* tokb = tokf + nCL;
  float* nstd = tokb + nCL;

  {
    int nthr = BB * LP;
    k_stats<<<(nthr + 255) / 256, 256, 0, stream>>>(x, mask, in_f, in_b, sbv);
  }
  {
    int tiles = BB * (CC / 16) * (LP / 48);   // 2*4*83 = 664 waves
    k_tok<<<tiles, 32, 0, stream>>>(w_fore, in_f, tokf);
    k_tok<<<tiles, 32, 0, stream>>>(w_back, in_b, tokb);
  }
  k_attn<<<BB * (LP / 16), 32, 0, stream>>>(tokf, tokb, sbv, rpb, nstd);
  {
    size_t n = (size_t)BB * CC * HH * WW;
    k_fold<<<(unsigned)((n + 255) / 256), 256, 0, stream>>>(
        x, mask, fg_gamma, fg_beta, bg_gamma, bg_beta, in_f, nstd, out);
  }
}